// GCNLayer_17016660427562
// MI455X (gfx1250) — compile-verified
//
#include <hip/hip_runtime.h>

// GCN layer: out = segment_sum(vals * (x@W)[cols], rows) + bias
// Phase 1 (HBM-bound, ~3.3us floor): sp = x @ W via V_WMMA_F32_16X16X4_F32.
//   fp32 tensor op = zero precision loss; GEMM is bandwidth-bound at 23.3 TB/s
//   so the modest K=4 WMMA rate is far above what memory can feed.
// Phase 2 (L2-bound, dominates): edge-parallel gather + global_atomic_add_f32;
//   sp (25.6MB) and out (25.6MB) both stay resident in the 192MB L2.

typedef __attribute__((ext_vector_type(2))) float v2f;
typedef __attribute__((ext_vector_type(8))) float v8f;

#define IN_F  128
#define OUT_F 64
// K-pair-interleaved W layout in LDS: element W[k][n] lives at
//   (k>>1)*PAIR_STRIDE + n*2 + (k&1)
// so the B fragment {W[k][n], W[k+1][n]} is one aligned ds_load_b64.
// PAIR_STRIDE=160 words => lanes 16-31 (pair-row p+1) hit banks 32..63 while
// lanes 0-15 hit banks 0..31: conflict-free.
#define PAIR_STRIDE 160

// ---------------------------------------------------------------------------
// Kernel 0: out[r][f] = bias[f]
// ---------------------------------------------------------------------------
__global__ void gcn_bias_init(const float* __restrict__ bias,
                              float* __restrict__ out, int total) {
    int gid = blockIdx.x * blockDim.x + threadIdx.x;
    if (gid < total) out[gid] = bias[gid & (OUT_F - 1)];
}

// ---------------------------------------------------------------------------
// Kernel 1: sp = x @ W  using v_wmma_f32_16x16x4_f32
//
// Fragment layouts (wave32, CDNA5 ISA):
//   A (16x4 f32):  lane L<16: row M=L,   K = k0 + {0,1}
//                  lane L>=16: row M=L-16, K = k0 + {2,3}
//   B (4x16 f32):  mirrored (lane -> N, VGPR/half -> K)
//   C/D (16x16):   VGPR i: lanes 0-15 -> M=i; lanes 16-31 -> M=i+8; N=lane&15
//
// Block = 256 threads = 8 waves; wave computes 16 rows x 64 cols
// (4 accumulators); block covers 128 rows. 128 WMMAs per wave.
// ---------------------------------------------------------------------------
__global__ __launch_bounds__(256) void gcn_gemm_wmma(
        const float* __restrict__ x, const float* __restrict__ w,
        float* __restrict__ sp, int n_nodes) {
    __shared__ float wlds[64 * PAIR_STRIDE];   // 40 KB of 320 KB/WGP

    const int tid = threadIdx.x;
    // Stage W (row-major [128,64]) into the pair-interleaved layout.
    for (int i = tid; i < (IN_F * OUT_F) / 4; i += 256) {
        float4 v = ((const float4*)w)[i];
        int e = i << 2;
        int k = e >> 6;        // W row   (0..127)
        int n = e & 63;        // W col base (multiple of 4, never crosses row)
        float* dst = wlds + (k >> 1) * PAIR_STRIDE + (k & 1);
        dst[(n + 0) * 2] = v.x;
        dst[(n + 1) * 2] = v.y;
        dst[(n + 2) * 2] = v.z;
        dst[(n + 3) * 2] = v.w;
    }
    __syncthreads();

    const int wave  = tid >> 5;
    const int lane  = tid & 31;
    const int m     = lane & 15;          // row-in-tile (A) / col-in-tile (B,D)
    const int half  = lane >> 4;          // 0 | 1
    const int koff  = half << 1;          // 0 | 2

    const int row_base = (blockIdx.x * 8 + wave) * 16;
    int row = row_base + m;
    if (row >= n_nodes) row = n_nodes - 1;   // clamp: EXEC stays all-1s for WMMA
    const float* __restrict__ xrow = x + (size_t)row * IN_F;

    v8f acc0 = {}, acc1 = {}, acc2 = {}, acc3 = {};

    #pragma unroll
    for (int k0 = 0; k0 < IN_F; k0 += 4) {
        // A fragment: 2 consecutive K values -> one 64-bit global load per lane
        v2f a = *(const v2f*)(xrow + k0 + koff);
        // B fragments: aligned ds_load_b64 each, conflict-free banking
        const float* wk = wlds + ((k0 >> 1) + half) * PAIR_STRIDE + (m << 1);
        v2f b0 = *(const v2f*)(wk + 0);
        v2f b1 = *(const v2f*)(wk + 32);
        v2f b2 = *(const v2f*)(wk + 64);
        v2f b3 = *(const v2f*)(wk + 96);
        // 8 args: (neg_a, A, neg_b, B, c_mod, C, reuse_a, reuse_b)
        acc0 = __builtin_amdgcn_wmma_f32_16x16x4_f32(false, a, false, b0, (short)0, acc0, false, false);
        acc1 = __builtin_amdgcn_wmma_f32_16x16x4_f32(false, a, false, b1, (short)0, acc1, false, false);
        acc2 = __builtin_amdgcn_wmma_f32_16x16x4_f32(false, a, false, b2, (short)0, acc2, false, false);
        acc3 = __builtin_amdgcn_wmma_f32_16x16x4_f32(false, a, false, b3, (short)0, acc3, false, false);
    }

    // D layout: VGPR i -> row = row_base + half*8 + i, col = nt*16 + m
    const int r0 = row_base + (half << 3);
    #pragma unroll
    for (int i = 0; i < 8; ++i) {
        int r = r0 + i;
        if (r < n_nodes) {
            float* o = sp + (size_t)r * OUT_F + m;
            o[0]  = acc0[i];
            o[16] = acc1[i];
            o[32] = acc2[i];
            o[48] = acc3[i];
        }
    }
}

// ---------------------------------------------------------------------------
// Kernel 2: for each edge e, out[rows[e]][:] += vals[e] * sp[cols[e]][:]
// One wave per edge; each lane handles a float2 of features:
//   - 8B/lane coalesced gather from sp (L2-resident)
//   - 2x global_atomic_add_f32 per lane, contiguous per edge
// Halves wave count & index-load redundancy vs 1 feature/lane.
// ---------------------------------------------------------------------------
__global__ void gcn_edge_agg(const float* __restrict__ sp,
                             const int* __restrict__ rows,
                             const int* __restrict__ cols,
                             const float* __restrict__ vals,
                             float* __restrict__ out, int n_edges) {
    long long gid = (long long)blockIdx.x * blockDim.x + threadIdx.x;
    int edge = (int)(gid >> 5);                 // 32 lanes per edge
    int f2   = ((int)gid & 31) << 1;            // feature pair base
    if (edge < n_edges) {
        int   c = cols[edge];
        int   r = rows[edge];
        float v = vals[edge];
        v2f s = *(const v2f*)(sp + (size_t)c * OUT_F + f2);
        float* o = out + (size_t)r * OUT_F + f2;
        atomicAdd(o,     v * s.x);
        atomicAdd(o + 1, v * s.y);
    }
}

// ---------------------------------------------------------------------------
extern "C" void kernel_launch(void* const* d_in, const int* in_sizes, int n_in,
                              void* d_out, int out_size, void* d_ws, size_t ws_size,
                              hipStream_t stream) {
    const float* x    = (const float*)d_in[0];   // [N, 128]
    const float* w    = (const float*)d_in[1];   // [128, 64]
    const float* bias = (const float*)d_in[2];   // [64]
    const int*   rows = (const int*)  d_in[3];   // [E]
    const int*   cols = (const int*)  d_in[4];   // [E]
    const float* vals = (const float*)d_in[5];   // [E]
    float*       out  = (float*)d_out;           // [N, 64]

    const int n_nodes = in_sizes[0] / IN_F;
    const int n_edges = in_sizes[3];

    // Scratch: sp = x @ W, n_nodes*64 floats (25.6 MB) in d_ws.
    float* sp = (float*)d_ws;

    // Phase 0: out = bias (row-broadcast); aggregation accumulates on top.
    const int total_out = n_nodes * OUT_F;
    gcn_bias_init<<<(total_out + 255) / 256, 256, 0, stream>>>(bias, out, total_out);

    // Phase 1: WMMA GEMM. 128 rows per block.
    const int gemm_blocks = (n_nodes + 127) / 128;
    gcn_gemm_wmma<<<gemm_blocks, 256, 0, stream>>>(x, w, sp, n_nodes);

    // Phase 2: edge aggregation, one wave per edge, float2 per lane.
    const long long total_agg = (long long)n_edges * 32;
    const int agg_blocks = (int)((total_agg + 255) / 256);
    gcn_edge_agg<<<agg_blocks, 256, 0, stream>>>(sp, rows, cols, vals, out, n_edges);
}